// H2GCNConv_824633721275
// MI455X (gfx1250) — compile-verified
//
#include <hip/hip_runtime.h>
#include <cstdint>

#define D_FEAT      128
#define OUT_STRIDE  256      // concat(x1, x2)
#define NTHREADS    256      // 8 waves (wave32)
#define WAVES_PB    8

// ---- CDNA5 async global->LDS helpers ---------------------------------------

__device__ __forceinline__ uint32_t lds_addr_of(const void* p) {
    // Generic LDS pointer: low 32 bits are the LDS byte offset.
    return (uint32_t)(uintptr_t)p;
}
__device__ __forceinline__ void async_stage_b64(uint32_t lds, const void* g) {
    unsigned long long ga = (unsigned long long)(uintptr_t)g;
    asm volatile("global_load_async_to_lds_b64 %0, %1, off"
                 :: "v"(lds), "v"(ga) : "memory");
}
__device__ __forceinline__ void wait_async_le1() { asm volatile("s_wait_asynccnt 0x1" ::: "memory"); }
__device__ __forceinline__ void wait_async_le0() { asm volatile("s_wait_asynccnt 0x0" ::: "memory"); }

// ---- Pass 1: per-row degree histogram --------------------------------------

__global__ __launch_bounds__(NTHREADS)
void count_kernel(const int* __restrict__ row, long long nE, int* __restrict__ counts) {
    long long i = (long long)blockIdx.x * NTHREADS + threadIdx.x;
    const long long stride = (long long)gridDim.x * NTHREADS;
    for (; i < nE; i += stride) atomicAdd(&counts[row[i]], 1);
}

// ---- Pass 2: exclusive scan (single block; n=50000 => microseconds) --------

__global__ __launch_bounds__(NTHREADS)
void scan_kernel(const int* __restrict__ counts,
                 int* __restrict__ offs, int* __restrict__ cursor, int n) {
    __shared__ int sdata[NTHREADS];
    __shared__ int carry_s;
    if (threadIdx.x == 0) carry_s = 0;
    __syncthreads();
    for (int base = 0; base < n; base += NTHREADS) {
        int i = base + (int)threadIdx.x;
        int v = (i < n) ? counts[i] : 0;
        sdata[threadIdx.x] = v;
        __syncthreads();
        for (int d = 1; d < NTHREADS; d <<= 1) {
            int t = (threadIdx.x >= (unsigned)d) ? sdata[threadIdx.x - d] : 0;
            __syncthreads();
            sdata[threadIdx.x] += t;
            __syncthreads();
        }
        int incl  = sdata[threadIdx.x];
        int carry = carry_s;
        if (i < n) { int ex = carry + incl - v; offs[i] = ex; cursor[i] = ex; }
        __syncthreads();
        if (threadIdx.x == NTHREADS - 1) carry_s = carry + incl;
        __syncthreads();
    }
    if (threadIdx.x == 0) offs[n] = carry_s;
}

// ---- Pass 3: bin edges into CSR order as packed (col, valbits) pairs -------

__global__ __launch_bounds__(NTHREADS)
void bin_kernel(const int* __restrict__ row, const int* __restrict__ col,
                const float* __restrict__ vals, long long nE,
                int* __restrict__ cursor, int2* __restrict__ pairs) {
    long long i = (long long)blockIdx.x * NTHREADS + threadIdx.x;
    const long long stride = (long long)gridDim.x * NTHREADS;
    for (; i < nE; i += stride) {
        int r = row[i];
        int p = atomicAdd(&cursor[r], 1);
        pairs[p] = make_int2(col[i], __float_as_int(vals[i]));
    }
}

// ---- Pass 4: CSR SpMM, one wave per output row, zero f32 atomics ----------
// Edge chunks are async-copied global->LDS (b64 pairs), double-buffered
// per-wave (no barriers: each wave owns its LDS slot), scalarized via
// v_readlane so gathers/stores use scalar (saddr) bases. Two accumulator
// sets (even/odd edges) halve the v_fmac dependency chain.

__global__ __launch_bounds__(NTHREADS)
void csr_spmm_kernel(const float* __restrict__ x,
                     const int*   __restrict__ offs,
                     const int2*  __restrict__ pairs,
                     float*       __restrict__ out,
                     int nRows, int hopOff) {
    __shared__ int2 recs[WAVES_PB][2][32];
    const int wave = (int)(threadIdx.x >> 5);
    const int lane = (int)(threadIdx.x & 31);

    for (int r = blockIdx.x * WAVES_PB + wave; r < nRows; r += gridDim.x * WAVES_PB) {
        const int start = __builtin_amdgcn_readfirstlane(offs[r]);
        const int end   = __builtin_amdgcn_readfirstlane(offs[r + 1]);
        const int deg   = end - start;
        float4 acc0 = make_float4(0.f, 0.f, 0.f, 0.f);
        float4 acc1 = make_float4(0.f, 0.f, 0.f, 0.f);

        if (deg > 0) {
            const int nChunks = (deg + 31) >> 5;
            {   // stage chunk 0 -> buffer 0 (clamped tail duplicates unused)
                int e = start + lane; if (e > end - 1) e = end - 1;
                async_stage_b64(lds_addr_of(&recs[wave][0][lane]), pairs + e);
            }
            for (int ch = 0; ch < nChunks; ++ch) {
                if (ch + 1 < nChunks) {
                    int e = start + ((ch + 1) << 5) + lane; if (e > end - 1) e = end - 1;
                    async_stage_b64(lds_addr_of(&recs[wave][(ch + 1) & 1][lane]), pairs + e);
                    wait_async_le1();           // current chunk landed; next in flight
                } else {
                    wait_async_le0();
                }
                int2 my = recs[wave][ch & 1][lane];           // ds_load_b64
                {   // warm L2/WGP$ for this lane's gather row (512B = prefetch x2)
                    const char* p = (const char*)(x + (size_t)(uint32_t)my.x * D_FEAT);
                    __builtin_prefetch(p, 0, 3);
                    __builtin_prefetch(p + 256, 0, 3);
                }

                int cnt = deg - (ch << 5); if (cnt > 32) cnt = 32;
                auto do_edge = [&](int i, float4& acc) {
                    int   c = __builtin_amdgcn_readlane(my.x, i);
                    float v = __int_as_float(__builtin_amdgcn_readlane(my.y, i));
                    const float4* xr =
                        reinterpret_cast<const float4*>(x + (size_t)(uint32_t)c * D_FEAT);
                    float4 m = xr[lane];                       // global_load_b128 (saddr)
                    acc.x = fmaf(m.x, v, acc.x);
                    acc.y = fmaf(m.y, v, acc.y);
                    acc.z = fmaf(m.z, v, acc.z);
                    acc.w = fmaf(m.w, v, acc.w);
                };
                if (cnt == 32) {
                    #pragma unroll
                    for (int i = 0; i < 32; i += 2) {          // two independent chains
                        do_edge(i,     acc0);
                        do_edge(i + 1, acc1);
                    }
                } else {
                    for (int i = 0; i < cnt; ++i) do_edge(i, (i & 1) ? acc1 : acc0);
                }
            }
        }
        float4 acc = make_float4(acc0.x + acc1.x, acc0.y + acc1.y,
                                 acc0.z + acc1.z, acc0.w + acc1.w);
        float4* o = reinterpret_cast<float4*>(out + (size_t)(uint32_t)r * OUT_STRIDE + hopOff);
        o[lane] = acc;                                          // global_store_b128
    }
}

// ---- Fallback: scatter with native f32 atomics (if workspace too small) ----

__global__ __launch_bounds__(NTHREADS)
void spmm_scatter_fallback(const float* __restrict__ x, const int* __restrict__ row,
                           const int* __restrict__ col, const float* __restrict__ vals,
                           float* __restrict__ out, long long nE, int hopOff) {
    const int lane = (int)(threadIdx.x & 31);
    long long e = (long long)blockIdx.x * WAVES_PB + (int)(threadIdx.x >> 5);
    const long long stride = (long long)gridDim.x * WAVES_PB;
    for (; e < nE; e += stride) {
        int   r = __builtin_amdgcn_readfirstlane(row[e]);
        int   c = __builtin_amdgcn_readfirstlane(col[e]);
        float v = __int_as_float(__builtin_amdgcn_readfirstlane(__float_as_int(vals[e])));
        const float4* xr = reinterpret_cast<const float4*>(x + (size_t)(uint32_t)c * D_FEAT);
        float4 m = xr[lane];
        float* o = out + (size_t)(uint32_t)r * OUT_STRIDE + hopOff + lane * 4;
        unsafeAtomicAdd(o + 0, m.x * v);
        unsafeAtomicAdd(o + 1, m.y * v);
        unsafeAtomicAdd(o + 2, m.z * v);
        unsafeAtomicAdd(o + 3, m.w * v);
    }
}

// ---- Launch -----------------------------------------------------------------

extern "C" void kernel_launch(void* const* d_in, const int* in_sizes, int n_in,
                              void* d_out, int out_size, void* d_ws, size_t ws_size,
                              hipStream_t stream) {
    const float* x     = (const float*)d_in[0];
    const int*   row1  = (const int*)  d_in[1];
    const int*   col1  = (const int*)  d_in[2];
    const float* vals1 = (const float*)d_in[3];
    const int*   row2  = (const int*)  d_in[4];
    const int*   col2  = (const int*)  d_in[5];
    const float* vals2 = (const float*)d_in[6];
    float*       out   = (float*)d_out;

    const long long E1 = in_sizes[1];
    const long long E2 = in_sizes[4];
    const long long N  = in_sizes[0] / D_FEAT;

    auto al = [](size_t v) { return (v + 255) & ~(size_t)255; };
    size_t o = 0;
    const size_t offs1_o = o; o += al(((size_t)N + 1) * 4);
    const size_t offs2_o = o; o += al(((size_t)N + 1) * 4);
    const size_t cur1_o  = o; o += al((size_t)N * 4);
    const size_t cur2_o  = o; o += al((size_t)N * 4);
    const size_t cnt1_o  = o; o += al((size_t)N * 4);
    const size_t cnt2_o  = o; o += al((size_t)N * 4);
    const size_t p1_o    = o; o += al((size_t)E1 * 8);
    const size_t p2_o    = o; o += al((size_t)E2 * 8);
    const size_t need    = o;

    auto edge_grid = [](long long nE) {
        long long g = (nE + NTHREADS - 1) / NTHREADS;
        return (unsigned)(g < 8192 ? g : 8192);
    };

    if (ws_size >= need) {
        char* w = (char*)d_ws;
        int*  offs1 = (int*)(w + offs1_o);  int* offs2 = (int*)(w + offs2_o);
        int*  cur1  = (int*)(w + cur1_o);   int* cur2  = (int*)(w + cur2_o);
        int*  cnt1  = (int*)(w + cnt1_o);   int* cnt2  = (int*)(w + cnt2_o);
        int2* pr1   = (int2*)(w + p1_o);    int2* pr2  = (int2*)(w + p2_o);

        hipMemsetAsync(cnt1, 0, (size_t)N * 4, stream);
        hipMemsetAsync(cnt2, 0, (size_t)N * 4, stream);

        count_kernel<<<edge_grid(E1), NTHREADS, 0, stream>>>(row1, E1, cnt1);
        count_kernel<<<edge_grid(E2), NTHREADS, 0, stream>>>(row2, E2, cnt2);

        scan_kernel<<<1, NTHREADS, 0, stream>>>(cnt1, offs1, cur1, (int)N);
        scan_kernel<<<1, NTHREADS, 0, stream>>>(cnt2, offs2, cur2, (int)N);

        bin_kernel<<<edge_grid(E1), NTHREADS, 0, stream>>>(row1, col1, vals1, E1, cur1, pr1);
        bin_kernel<<<edge_grid(E2), NTHREADS, 0, stream>>>(row2, col2, vals2, E2, cur2, pr2);

        const unsigned gridS = (unsigned)((N + WAVES_PB - 1) / WAVES_PB);
        csr_spmm_kernel<<<gridS, NTHREADS, 0, stream>>>(x, offs1, pr1, out, (int)N, 0);
        csr_spmm_kernel<<<gridS, NTHREADS, 0, stream>>>(x, offs2, pr2, out, (int)N, D_FEAT);
        // Every output element is stored by the two SpMM passes: no out memset.
    } else {
        hipMemsetAsync(d_out, 0, (size_t)out_size * sizeof(float), stream);
        auto wave_grid = [](long long nE) {
            long long g = (nE + WAVES_PB - 1) / WAVES_PB;
            return (unsigned)(g < 16384 ? g : 16384);
        };
        spmm_scatter_fallback<<<wave_grid(E1), NTHREADS, 0, stream>>>(x, row1, col1, vals1, out, E1, 0);
        spmm_scatter_fallback<<<wave_grid(E2), NTHREADS, 0, stream>>>(x, row2, col2, vals2, out, E2, D_FEAT);
    }
}